// TripletLoss_5746666242139
// MI455X (gfx1250) — compile-verified
//
#include <hip/hip_runtime.h>

typedef __attribute__((ext_vector_type(2))) float v2f;
typedef __attribute__((ext_vector_type(8))) float v8f;
typedef __attribute__((ext_vector_type(4))) unsigned int u32x4;
typedef __attribute__((ext_vector_type(8))) int i32x8;
typedef __attribute__((ext_vector_type(4))) int i32x4;

#define B_N      8192
#define D_K      256
#define MARGIN_F 0.3f
#define BM       32
#define BN       64
#define NPANELS  (B_N / BN)   // 128
#define LDSPAD   260          // 256 + 4 floats pad; TDM pad: 4 DWORDs per 256 DWORDs

__global__ void init_ws_kernel(float* ws) {
    if (threadIdx.x == 0) { ws[0] = 0.0f; ws[1] = 0.0f; }
}

// One wave (32 lanes) per row: sq[i] = sum_j x[i][j]^2
__global__ __launch_bounds__(256) void row_sqnorm_kernel(const float* __restrict__ x,
                                                         float* __restrict__ sq) {
    const int wid  = threadIdx.x >> 5;
    const int lane = threadIdx.x & 31;
    const int row  = blockIdx.x * 8 + wid;
    const float* r = x + (size_t)row * D_K;
    float s = 0.0f;
    #pragma unroll
    for (int j = 0; j < D_K / 32; ++j) {
        float v = r[lane + j * 32];
        s = fmaf(v, v, s);
    }
    #pragma unroll
    for (int m = 16; m >= 1; m >>= 1) s += __shfl_xor(s, m);
    if (lane == 0) sq[row] = s;
}

// TDM: DMA one BN x D_K f32 panel (row stride 256 fl) into LDS with 260-float padded rows.
__device__ __forceinline__ void tdm_load_panel(const float* gsrc, unsigned ldsByteOff) {
    unsigned long long ga = (unsigned long long)(uintptr_t)gsrc;
    u32x4 g0;
    g0.x = 1u;                                                   // count=1 (valid, user mode)
    g0.y = ldsByteOff;                                           // lds_addr (bytes)
    g0.z = (unsigned)(ga & 0xFFFFFFFFull);                       // global_addr[31:0]
    g0.w = (unsigned)((ga >> 32) & 0x1FFFFFFull) | (2u << 30);   // global_addr[56:32] | type=2
    i32x8 g1;
    g1[0] = (int)((2u << 16)      // data_size = 4 bytes
                | (1u << 20)      // pad_enable
                | (7u << 22)      // pad_interval: 256 DWORDs (one 1024B row)
                | (3u << 25));    // pad_amount: 4 DWORDs -> 1040B LDS row stride
    g1[1] = (int)(256u << 16);    // tensor_dim0 = 256 (low 16 bits at [63:48])
    g1[2] = (int)(8192u << 16);   // tensor_dim0 hi = 0; tensor_dim1 = 8192 (low 16 at [95:80])
    g1[3] = (int)(256u << 16);    // tensor_dim1 hi = 0; tile_dim0 = 256
    g1[4] = BN;                   // tile_dim1 = BN rows; tile_dim2 = 0 (2D)
    g1[5] = 256;                  // tensor_dim0_stride = 256 elements (low 32 of 48)
    g1[6] = 0;                    // stride hi, dim1_stride lo
    g1[7] = 0;
    i32x4 gz = {0, 0, 0, 0};
#if defined(__clang_major__) && (__clang_major__ >= 23)
    i32x8 gz8 = {0, 0, 0, 0, 0, 0, 0, 0};
    __builtin_amdgcn_tensor_load_to_lds(g0, g1, gz, gz, gz8, 0);
#else
    __builtin_amdgcn_tensor_load_to_lds(g0, g1, gz, gz, 0);
#endif
}

// Fused X*X^T (f32 WMMA) + hard positive/negative mining, TDM double-buffered B panels.
__global__ __launch_bounds__(256) void triplet_main_kernel(const float* __restrict__ x,
                                                           const int*   __restrict__ tgt,
                                                           const float* __restrict__ sq,
                                                           float*       __restrict__ ws) {
    __shared__ float As[BM * LDSPAD];
    __shared__ float Bs[2][BN * LDSPAD];
    __shared__ float redP[BM][BN / 16];
    __shared__ float redN[BM][BN / 16];

    const int tid      = threadIdx.x;
    const int wid      = tid >> 5;       // 8 waves
    const int lane     = tid & 31;
    const int half     = lane >> 4;
    const int lm       = lane & 15;
    const int rowTile  = wid >> 2;       // 0..1
    const int colTile  = wid & 3;        // 0..3
    const int blockRow = blockIdx.x * BM;

    // Stage the A strip (BM x D_K) once, plain loads (one-shot, L2-resident).
    const float4* g4 = (const float4*)x;
    for (int i = tid; i < BM * (D_K / 4); i += 256) {
        int r = i >> 6, c4 = i & 63;
        float4 v = g4[(size_t)(blockRow + r) * (D_K / 4) + c4];
        *(float4*)&As[r * LDSPAD + c4 * 4] = v;
    }

    // Wave 0 primes the TDM pipeline: panels 0 and 1 in flight.
    const unsigned ldsB0 = (unsigned)(size_t)(const void*)&Bs[0][0];
    const unsigned ldsB1 = (unsigned)(size_t)(const void*)&Bs[1][0];
    if (wid == 0) {
        tdm_load_panel(x, ldsB0);
        tdm_load_panel(x + (size_t)BN * D_K, ldsB1);
    }

    int trow[8];
    #pragma unroll
    for (int i = 0; i < 8; ++i)
        trow[i] = tgt[blockRow + rowTile * 16 + half * 8 + i];

    float pmax[8], nmin[8];
    #pragma unroll
    for (int i = 0; i < 8; ++i) { pmax[i] = -__builtin_inff(); nmin[i] = __builtin_inff(); }

    // ISA A layout 16x4 f32: lane L -> row L%16, K-pair (L/16)*2 (+vgpr). B symmetric.
    const float* aRow  = &As[(rowTile * 16 + lm) * LDSPAD + half * 2];
    const float* bRow0 = &Bs[0][(colTile * 16 + lm) * LDSPAD + half * 2];
    const float* bRow1 = &Bs[1][(colTile * 16 + lm) * LDSPAD + half * 2];

    for (int p = 0; p < NPANELS; ++p) {
        if (wid == 0) {  // panels complete in order: <=1 outstanding => panel p resident
            if (p + 1 < NPANELS) __builtin_amdgcn_s_wait_tensorcnt(1);
            else                 __builtin_amdgcn_s_wait_tensorcnt(0);
        }
        __syncthreads();   // releases TDM data; also covers As staging on p==0

        const float* bRow = (p & 1) ? bRow1 : bRow0;
        v8f acc = {};
        #pragma unroll 8
        for (int k = 0; k < D_K; k += 4) {
            v2f a2 = *(const v2f*)(aRow + k);
            v2f b2 = *(const v2f*)(bRow + k);
            acc = __builtin_amdgcn_wmma_f32_16x16x4_f32(
                false, a2, false, b2, (short)0, acc, false, false);
        }

        const int   colBase = p * BN;
        const int   colG    = colBase + colTile * 16 + lm;
        const float sqc     = sq[colG];
        const int   tcol    = tgt[colG];
        #pragma unroll
        for (int i = 0; i < 8; ++i) {
            float v   = fmaf(-2.0f, acc[i], sqc);   // sq_col - 2*dot; sq_row cancels
            bool  pos = (tcol == trow[i]);
            float pm = pmax[i], nm = nmin[i];
            pmax[i] = (pos  && v > pm) ? v : pm;
            nmin[i] = (!pos && v < nm) ? v : nm;
        }

        __syncthreads();   // all waves done reading buffer (p&1) before TDM overwrites it
        if (wid == 0 && p + 2 < NPANELS) {
            tdm_load_panel(x + (size_t)(p + 2) * BN * D_K, (p & 1) ? ldsB1 : ldsB0);
        }
    }

    // Butterfly-reduce across the 16 lanes sharing each row (masks 1..8 stay in-half).
    #pragma unroll
    for (int i = 0; i < 8; ++i) {
        #pragma unroll
        for (int m = 1; m <= 8; m <<= 1) {
            pmax[i] = fmaxf(pmax[i], __shfl_xor(pmax[i], m));
            nmin[i] = fminf(nmin[i], __shfl_xor(nmin[i], m));
        }
    }
    if (lm == 0) {
        #pragma unroll
        for (int i = 0; i < 8; ++i) {
            int r = rowTile * 16 + half * 8 + i;
            redP[r][colTile] = pmax[i];
            redN[r][colTile] = nmin[i];
        }
    }
    __syncthreads();

    if (tid < BM) {  // wave 0 finishes: one lane per row
        float pm = redP[tid][0], nm = redN[tid][0];
        #pragma unroll
        for (int g = 1; g < BN / 16; ++g) {
            pm = fmaxf(pm, redP[tid][g]);
            nm = fminf(nm, redN[tid][g]);
        }
        float loss = fmaxf(0.0f, pm - nm + MARGIN_F);
        float prec = (nm > pm) ? 1.0f : 0.0f;
        #pragma unroll
        for (int m = 16; m >= 1; m >>= 1) {
            loss += __shfl_xor(loss, m);
            prec += __shfl_xor(prec, m);
        }
        if (tid == 0) {
            atomicAdd(&ws[0], loss);
            atomicAdd(&ws[1], prec);
        }
    }
}

__global__ void finalize_kernel(const float* __restrict__ ws, float* __restrict__ out) {
    if (threadIdx.x == 0) {
        const float invB = 1.0f / (float)B_N;
        out[0] = ws[0] * invB;
        out[1] = ws[1] * invB;
    }
}

extern "C" void kernel_launch(void* const* d_in, const int* in_sizes, int n_in,
                              void* d_out, int out_size, void* d_ws, size_t ws_size,
                              hipStream_t stream) {
    (void)in_sizes; (void)n_in; (void)out_size; (void)ws_size;
    const float* x   = (const float*)d_in[0];
    const int*   tgt = (const int*)d_in[1];
    float*       ws  = (float*)d_ws;        // [0]=loss sum, [1]=prec sum, [2..]=sq norms
    float*       sq  = ws + 2;
    float*       out = (float*)d_out;

    init_ws_kernel<<<1, 32, 0, stream>>>(ws);
    row_sqnorm_kernel<<<B_N / 8, 256, 0, stream>>>(x, sq);
    triplet_main_kernel<<<B_N / BM, 256, 0, stream>>>(x, tgt, sq, ws);
    finalize_kernel<<<1, 32, 0, stream>>>(ws, out);
}